// WassersteinSelfAttentionModel_28415503630807
// MI455X (gfx1250) — compile-verified
//
#include <hip/hip_runtime.h>

typedef __attribute__((ext_vector_type(2))) float v2f;
typedef __attribute__((ext_vector_type(4))) float v4f;
typedef __attribute__((ext_vector_type(8))) float v8f;

#define B_ 4
#define S_ 2048
#define D_ 512
#define H_ 512
#define NCHUNK 64
#define ROWS_PER_CHUNK (S_ / NCHUNK) /* 32 */

// ws layout (in floats):
//   [0, NCHUNK*B_*D_)           : chunk partial sums, [chunk][b][d]
//   [WS_XS, WS_XS + B_*D_)      : xs[b][d] = (1/S) * colsum(x)
//   [WS_M,  WS_M  + B_*H_)      : m[b][h]  = xs @ Wv + bv
#define WS_XS (NCHUNK * B_ * D_)
#define WS_M (WS_XS + B_ * D_)

// Stage 1: per-chunk column sums of x over S. 16 MB streamed once.
__global__ void colsum_partial(const float* __restrict__ x, float* __restrict__ ws) {
    int wg = blockIdx.x;             // 0 .. B_*NCHUNK-1
    int b = wg / NCHUNK;
    int c = wg % NCHUNK;
    int t = threadIdx.x;             // 0..127, each owns 4 consecutive d
    const float* xp = x + ((size_t)(b * S_ + c * ROWS_PER_CHUNK)) * D_ + t * 4;
    v4f acc = {0.f, 0.f, 0.f, 0.f};
#pragma unroll 4
    for (int j = 0; j < ROWS_PER_CHUNK; ++j) {
        v4f v = *(const v4f*)(xp + (size_t)j * D_);
        acc += v;
    }
    *(v4f*)(ws + ((size_t)(c * B_ + b)) * D_ + t * 4) = acc;
}

// Stage 2: fold the NCHUNK partials, pre-scale by 1/S.
__global__ void reduce_chunks(float* __restrict__ ws) {
    int idx = blockIdx.x * blockDim.x + threadIdx.x;  // 0..2047
    int b = idx >> 9;
    int d = idx & (D_ - 1);
    const float* p = ws + b * D_ + d;
    float s = 0.f;
#pragma unroll 8
    for (int c = 0; c < NCHUNK; ++c) s += p[(size_t)c * (B_ * D_)];
    ws[WS_XS + idx] = s * (1.0f / (float)S_);
}

// Stage 3: m[4,512] = xs[4,512] @ Wv[512,512] + bv via V_WMMA_F32_16X16X4_F32.
// The 4x512 xs is staged into LDS zero-padded to 16x512 (32 KB), so the
// K-loop needs no predication: A fragments are plain ds_load_b64, EXEC stays
// all-1s (WMMA requirement), and there are no exec-mask branches.
// A layout (16x4 f32): lanes 0-15 M=lane K={0(v0),1(v1)}; lanes 16-31 M=lane-16 K={2,3}.
// B layout (4x16 f32): lanes 0-15 N=lane K={0(v0),1(v1)}; lanes 16-31 N=lane-16 K={2,3}.
// C layout: VGPR r, lanes 0-15 -> (M=r, N=lane). We keep rows M=0..3.
__global__ void mean_gemm_wmma(const float* __restrict__ Wv,
                               const float* __restrict__ bv,
                               float* __restrict__ ws) {
    const float* xs = ws + WS_XS;
    float* mout = ws + WS_M;

    __shared__ float lA[16 * D_];  // zero-padded A, rows 4..15 = 0

    // Cooperative fill: rows 0..3 from xs, rows 4..15 zero.
    for (int i = threadIdx.x; i < 16 * D_; i += blockDim.x)
        lA[i] = (i < B_ * D_) ? xs[i] : 0.f;
    __syncthreads();

    int lane = threadIdx.x & 31;
    int wave = threadIdx.x >> 5;
    int tile = blockIdx.x * 4 + wave;  // 0..31 -> N columns [16*tile, 16*tile+16)
    int nloc = lane & 15;
    int n = tile * 16 + nloc;
    int kh = (lane >> 4) << 1;   // 0 for lanes 0-15, 2 for lanes 16-31
    int m = lane & 15;           // padded M row this lane carries

    const float* lAm = lA + m * D_ + kh;  // this lane's A row, K-half adjusted

    v8f c0 = {0.f, 0.f, 0.f, 0.f, 0.f, 0.f, 0.f, 0.f};
    v8f c1 = {0.f, 0.f, 0.f, 0.f, 0.f, 0.f, 0.f, 0.f};

    for (int k = 0; k < D_; k += 8) {
        v2f a0 = *(const v2f*)(lAm + k);       // K = k+kh, k+kh+1   (ds_load_b64)
        v2f a1 = *(const v2f*)(lAm + k + 4);   // K = k+4+kh, k+5+kh (ds_load_b64)
        v2f b0, b1;
        b0.x = Wv[(size_t)(k + kh) * H_ + n];
        b0.y = Wv[(size_t)(k + kh + 1) * H_ + n];
        b1.x = Wv[(size_t)(k + 4 + kh) * H_ + n];
        b1.y = Wv[(size_t)(k + 5 + kh) * H_ + n];
        c0 = __builtin_amdgcn_wmma_f32_16x16x4_f32(false, a0, false, b0,
                                                   (short)0, c0, false, false);
        c1 = __builtin_amdgcn_wmma_f32_16x16x4_f32(false, a1, false, b1,
                                                   (short)0, c1, false, false);
    }
    v8f c = c0 + c1;

    if (lane < 16) {
#pragma unroll
        for (int r = 0; r < B_; ++r)
            mout[r * H_ + tile * 16 + nloc] = c[r] + bv[tile * 16 + nloc];
    }
}

// Stage 4: broadcast m[b,:] across all S positions (16 MB of float4 stores).
__global__ void broadcast_out(const float* __restrict__ ws, float* __restrict__ out) {
    const float* mout = ws + WS_M;
    size_t gid = (size_t)blockIdx.x * blockDim.x + threadIdx.x;  // 0 .. B_*S_*H_/4-1
    int b = (int)(gid / ((size_t)S_ * H_ / 4));
    int h4 = (int)(gid & (H_ / 4 - 1));
    v4f val = *(const v4f*)(mout + b * H_ + h4 * 4);
    *((v4f*)out + gid) = val;
}

extern "C" void kernel_launch(void* const* d_in, const int* in_sizes, int n_in,
                              void* d_out, int out_size, void* d_ws, size_t ws_size,
                              hipStream_t stream) {
    // setup_inputs order: x, Wq, bq, Wk, bk, Wv, bv
    const float* x = (const float*)d_in[0];
    const float* Wv = (const float*)d_in[5];
    const float* bv = (const float*)d_in[6];
    float* out = (float*)d_out;
    float* ws = (float*)d_ws;

    colsum_partial<<<B_ * NCHUNK, 128, 0, stream>>>(x, ws);
    reduce_chunks<<<(B_ * D_) / 256, 256, 0, stream>>>(ws);
    mean_gemm_wmma<<<8, 128, 0, stream>>>(Wv, bv, ws);
    broadcast_out<<<(B_ * S_ * H_ / 4) / 256, 256, 0, stream>>>(ws, out);
}